// SAModule_42906723287328
// MI455X (gfx1250) — compile-verified
//
#include <hip/hip_runtime.h>

#define BATCH 8
#define NPB   8192
#define FDIM  64
#define MQ    2048
#define KNBR  32
#define CDIM  128
#define R2    0.01f      // R = 0.1

typedef float v2f __attribute__((ext_vector_type(2)));
typedef float v8f __attribute__((ext_vector_type(8)));

// --------------------------------------------------------------------------
// Kernel 1: farthest point sampling, one workgroup per cloud.
// Cloud positions resident in LDS (96KB of the WGP's 320KB).
// dmin kept in registers (8 points per thread). Two barriers per iteration.
// --------------------------------------------------------------------------
__global__ __launch_bounds__(1024) void fps_kernel(
    const float* __restrict__ pos, int* __restrict__ samp,
    float* __restrict__ qpos_out, int* __restrict__ batch_out,
    int* __restrict__ idx_out)
{
  extern __shared__ float sm[];
  float* px = sm;
  float* py = sm + NPB;
  float* pz = sm + 2 * NPB;
  float* rv = sm + 3 * NPB;                 // 32 wave maxima
  int*   ri = (int*)(sm + 3 * NPB + 32);    // 32 wave argmax
  int*   bcast = (int*)(sm + 3 * NPB + 64); // winner broadcast

  const int b   = blockIdx.x;
  const int tid = threadIdx.x;
  const int lane = tid & 31;
  const int wave = tid >> 5;
  const float* posb = pos + (size_t)b * NPB * 3;

  for (int i = tid; i < NPB; i += 1024) {
    px[i] = posb[i * 3 + 0];
    py[i] = posb[i * 3 + 1];
    pz[i] = posb[i * 3 + 2];
  }
  __syncthreads();

  float dmin[8];
#pragma unroll
  for (int t = 0; t < 8; ++t) dmin[t] = 3.0e38f;

  int last = 0;
  for (int it = 0; it < MQ; ++it) {
    if (tid == 0) {
      samp[b * MQ + it] = last;
      qpos_out[(b * MQ + it) * 3 + 0] = px[last];
      qpos_out[(b * MQ + it) * 3 + 1] = py[last];
      qpos_out[(b * MQ + it) * 3 + 2] = pz[last];
      batch_out[b * MQ + it] = b;
      idx_out[b * MQ + it] = b * NPB + last;
    }
    const float lx = px[last], ly = py[last], lz = pz[last];

    float bv = -1.0f;
    int   bi = 0;
#pragma unroll
    for (int t = 0; t < 8; ++t) {
      const int i = t * 1024 + tid;
      const float dx = px[i] - lx, dy = py[i] - ly, dz = pz[i] - lz;
      const float d = dx * dx + dy * dy + dz * dz;
      const float dm = fminf(dmin[t], d);
      dmin[t] = dm;
      if (dm > bv) { bv = dm; bi = i; }
    }
    // wave32 argmax (prefer lowest index on ties, like jnp.argmax)
#pragma unroll
    for (int off = 16; off > 0; off >>= 1) {
      const float ov = __shfl_xor(bv, off, 32);
      const int   oi = __shfl_xor(bi, off, 32);
      if (ov > bv || (ov == bv && oi < bi)) { bv = ov; bi = oi; }
    }
    if (lane == 0) { rv[wave] = bv; ri[wave] = bi; }
    __syncthreads();
    if (wave == 0) {
      float v = rv[lane];
      int   i2 = ri[lane];
#pragma unroll
      for (int off = 16; off > 0; off >>= 1) {
        const float ov = __shfl_xor(v, off, 32);
        const int   oi = __shfl_xor(i2, off, 32);
        if (ov > v || (ov == v && oi < i2)) { v = ov; i2 = oi; }
      }
      if (lane == 0) bcast[0] = i2;
    }
    __syncthreads();
    last = bcast[0];
  }
}

// --------------------------------------------------------------------------
// Kernel 2: radius ball query, K nearest within R. One thread per query,
// candidate positions tiled through LDS. Register-resident top-32 list
// (fully unrolled replace-max so it never spills to scratch). Unused slots
// are filled with the self index -> duplicates, harmless under max-pool.
// --------------------------------------------------------------------------
__global__ __launch_bounds__(256) void ballq_kernel(
    const float* __restrict__ pos, const float* __restrict__ qpos,
    const int* __restrict__ samp, int* __restrict__ nbr)
{
  __shared__ float tpx[1024], tpy[1024], tpz[1024];
  const int tid = threadIdx.x;
  const int q = blockIdx.x * 256 + tid;
  const int b = q / MQ;
  const float qx = qpos[q * 3 + 0];
  const float qy = qpos[q * 3 + 1];
  const float qz = qpos[q * 3 + 2];
  const int self = samp[q];

  float kd[KNBR];
  int   kn[KNBR];
#pragma unroll
  for (int s = 0; s < KNBR; ++s) { kd[s] = 3.0e38f; kn[s] = self; }
  float curmax = 3.0e38f;
  int   maxslot = 0;

  const float* posb = pos + (size_t)b * NPB * 3;
  for (int tile = 0; tile < NPB / 1024; ++tile) {
    __syncthreads();
    for (int j = tid; j < 1024; j += 256) {
      const int p = tile * 1024 + j;
      tpx[j] = posb[p * 3 + 0];
      tpy[j] = posb[p * 3 + 1];
      tpz[j] = posb[p * 3 + 2];
    }
    __syncthreads();
    for (int j = 0; j < 1024; ++j) {
      const float dx = tpx[j] - qx, dy = tpy[j] - qy, dz = tpz[j] - qz;
      const float d2 = dx * dx + dy * dy + dz * dz;
      if (d2 <= R2 && d2 < curmax) {
#pragma unroll
        for (int s = 0; s < KNBR; ++s)
          if (s == maxslot) { kd[s] = d2; kn[s] = tile * 1024 + j; }
        curmax = -1.0f;
#pragma unroll
        for (int s = 0; s < KNBR; ++s)
          if (kd[s] > curmax) { curmax = kd[s]; maxslot = s; }
      }
    }
  }
#pragma unroll
  for (int s = 0; s < KNBR; ++s) nbr[q * KNBR + s] = kn[s];
}

// --------------------------------------------------------------------------
// Kernel 3: gather + PointNet conv via fp32 WMMA + ReLU + neighbor max-pool.
// 8 queries per 128-thread block (amortizes the 67x128 W1 staging 8x).
// The neighbor-feature gather uses CDNA5 GLOBAL_LOAD_ASYNC_TO_LDS_B128
// (ASYNCcnt-tracked, per-lane global->LDS), double-buffered so the next
// query's gather overlaps the current query's 68-step WMMA loop.
// A tile: 32x72 (64 feats + 3 relpos + pad; pitch 72 keeps B128 16B-aligned).
// --------------------------------------------------------------------------
#define QPB    8
#define APITCH 72
#define WPITCH 144

__global__ __launch_bounds__(128) void conv_kernel(
    const float* __restrict__ x, const float* __restrict__ pos,
    const float* __restrict__ W1, const float* __restrict__ b1,
    const float* __restrict__ qpos, const int* __restrict__ nbr,
    float* __restrict__ out)
{
  __shared__ __attribute__((aligned(16))) float A[2][KNBR][APITCH];
  __shared__ __attribute__((aligned(16))) float Wl[68][WPITCH];
  __shared__ float b1l[CDIM];

  const int q0 = blockIdx.x * QPB;
  const int b = q0 / MQ;               // QPB | MQ -> one cloud per block
  const int tid = threadIdx.x;
  const int lane = tid & 31;
  const int wave = tid >> 5;

  // stage W1 (67x128) vectorized; zero pad row 67 so K=68 is a clean 17x4
  {
    const float4* W4 = (const float4*)W1;
    const float4 z4 = make_float4(0.f, 0.f, 0.f, 0.f);
    for (int i = tid; i < 68 * 32; i += 128) {
      const int r = i >> 5, c = i & 31;
      *(float4*)&Wl[r][c * 4] = (r < 67) ? W4[r * 32 + c] : z4;
    }
    if (tid < CDIM) b1l[tid] = b1[tid];
  }

  const uint64_t xbase = (uint64_t)(size_t)x;
  const int n  = tid >> 2;   // neighbor row 0..31
  const int c4 = tid & 3;    // 64B chunk of the 256B feature row

  // async gather of query (q0+qi)'s 32 neighbor rows into A[buf]
  auto gather = [&](int qi, int buf) {
    const int q = q0 + qi;
    const int gl = nbr[q * KNBR + n];
    const uint32_t ldso = (uint32_t)(size_t)&A[buf][n][c4 * 16];
    const uint32_t goff =
        (uint32_t)((((uint32_t)(b * NPB + gl)) * FDIM + c4 * 16) * 4);
    asm volatile(
        "global_load_async_to_lds_b128 %0, %1, %2 offset:0\n\t"
        "global_load_async_to_lds_b128 %0, %1, %2 offset:16\n\t"
        "global_load_async_to_lds_b128 %0, %1, %2 offset:32\n\t"
        "global_load_async_to_lds_b128 %0, %1, %2 offset:48"
        :: "v"(ldso), "v"(goff), "s"(xbase) : "memory");
    if (c4 == 0) {
      const float* pr = pos + ((size_t)b * NPB + gl) * 3;
      A[buf][n][FDIM + 0] = pr[0] - qpos[q * 3 + 0];
      A[buf][n][FDIM + 1] = pr[1] - qpos[q * 3 + 1];
      A[buf][n][FDIM + 2] = pr[2] - qpos[q * 3 + 2];
      A[buf][n][FDIM + 3] = 0.0f;
    }
  };

  gather(0, 0);

  const int n16 = lane & 15;
  const int khalf = (lane >> 4) * 2;  // lanes 0-15: K{0,1}; 16-31: K{2,3}

  for (int qi = 0; qi < QPB; ++qi) {
    const int buf = qi & 1;
    asm volatile("s_wait_asynccnt 0" ::: "memory");
    __syncthreads();                      // A[buf] + Wl visible to all waves
    if (qi + 1 < QPB) gather(qi + 1, buf ^ 1);  // overlap with WMMA below

    const int q = q0 + qi;
    for (int cti = 0; cti < 2; ++cti) {
      const int ct = wave * 2 + cti;
      const int col = ct * 16 + n16;
      v8f acc0 = {0.f, 0.f, 0.f, 0.f, 0.f, 0.f, 0.f, 0.f};
      v8f acc1 = {0.f, 0.f, 0.f, 0.f, 0.f, 0.f, 0.f, 0.f};
#pragma unroll
      for (int kk = 0; kk < 17; ++kk) {
        const int kb = kk * 4 + khalf;
        v2f bf, a0, a1;
        bf.x = Wl[kb + 0][col];
        bf.y = Wl[kb + 1][col];
        a0.x = A[buf][n16][kb + 0];
        a0.y = A[buf][n16][kb + 1];
        a1.x = A[buf][16 + n16][kb + 0];
        a1.y = A[buf][16 + n16][kb + 1];
        acc0 = __builtin_amdgcn_wmma_f32_16x16x4_f32(
            false, a0, false, bf, (short)0, acc0, false, false);
        acc1 = __builtin_amdgcn_wmma_f32_16x16x4_f32(
            false, a1, false, bf, (short)0, acc1, false, false);
      }
      // bias + relu + max over the 32 neighbor rows of this column tile
      const float bias = b1l[col];
      float mx = 0.0f;
#pragma unroll
      for (int r = 0; r < 8; ++r) {
        mx = fmaxf(mx, fmaxf(acc0[r] + bias, 0.0f));
        mx = fmaxf(mx, fmaxf(acc1[r] + bias, 0.0f));
      }
      mx = fmaxf(mx, __shfl_xor(mx, 16, 32));  // combine lane halves (rows)
      if (lane < 16) out[(size_t)q * CDIM + col] = mx;
    }
  }
}

// --------------------------------------------------------------------------
extern "C" void kernel_launch(void* const* d_in, const int* in_sizes, int n_in,
                              void* d_out, int out_size, void* d_ws, size_t ws_size,
                              hipStream_t stream)
{
  const float* x    = (const float*)d_in[0];
  const float* pos  = (const float*)d_in[1];
  /* const int* batch = (const int*)d_in[2];  // implied by cloud layout */
  const float* W1   = (const float*)d_in[3];
  const float* b1   = (const float*)d_in[4];

  // d_out = concat(out[B*M,C], qpos[B*M,3], batch[B*M] i32, idx[B*M] i32)
  float* out    = (float*)d_out;
  float* qposo  = out + (size_t)BATCH * MQ * CDIM;
  int*   batcho = (int*)(qposo + (size_t)BATCH * MQ * 3);
  int*   idxo   = batcho + BATCH * MQ;

  int* samp = (int*)d_ws;              // [B*M]
  int* nbr  = samp + BATCH * MQ;       // [B*M*K]

  const size_t shb = (size_t)(3 * NPB + 80) * sizeof(float);
  fps_kernel<<<BATCH, 1024, shb, stream>>>(pos, samp, qposo, batcho, idxo);
  ballq_kernel<<<(BATCH * MQ) / 256, 256, 0, stream>>>(pos, qposo, samp, nbr);
  conv_kernel<<<(BATCH * MQ) / QPB, 128, 0, stream>>>(x, pos, W1, b1, qposo, nbr, out);
}